// DBGNN_26972394619794
// MI455X (gfx1250) — compile-verified
//
#include <hip/hip_runtime.h>
#include <math.h>

typedef __attribute__((ext_vector_type(2))) float v2f;
typedef __attribute__((ext_vector_type(8))) float v8f;

#define THREADS 256

// ---------------------------------------------------------------------------
// deg[v] = 1 (self loop)
__global__ void k_init_deg(float* __restrict__ deg, int n) {
    int i = blockIdx.x * blockDim.x + threadIdx.x;
    if (i < n) deg[i] = 1.0f;
}

// deg[dst[e]] += ew[e]
__global__ void k_deg_edges(float* __restrict__ deg,
                            const int* __restrict__ dst,
                            const float* __restrict__ ew, int e_cnt) {
    int e = blockIdx.x * blockDim.x + threadIdx.x;
    if (e < e_cnt) atomicAdd(&deg[dst[e]], ew[e]);
}

// in place: deg -> rsqrt(deg) (deg>0 guaranteed by self loop, keep guard)
__global__ void k_dinv(float* __restrict__ deg, int n) {
    int i = blockIdx.x * blockDim.x + threadIdx.x;
    if (i < n) {
        float d = deg[i];
        deg[i] = (d > 0.0f) ? rsqrtf(d) : 0.0f;
    }
}

// ---------------------------------------------------------------------------
// Y[M,NOUT] = X[M,128] @ W[128,NOUT] (+bias), fp32 WMMA 16x16x4.
// One wave computes a 16 x NOUT stripe. M tiles of 16; Mtiles check is
// wave-uniform so EXEC is all-ones around every v_wmma.
template <int NOUT, bool BIAS>
__global__ __launch_bounds__(THREADS) void k_wmma_gemm_k128(
    const float* __restrict__ X, const float* __restrict__ W,
    const float* __restrict__ bias, float* __restrict__ Y, int mtiles) {
    constexpr int K = 128;
    constexpr int NT = NOUT / 16;
    const int lane = threadIdx.x & 31;
    const int wv = threadIdx.x >> 5;
    const int mtile = blockIdx.x * (THREADS / 32) + wv;
    if (mtile >= mtiles) return;

    const int l15 = lane & 15;          // M (for A) / N (for B,C)
    const int half2 = (lane >> 4) * 2;  // K sub-select: 0 or 2
    const float* xrow = X + (size_t)(mtile * 16 + l15) * K;

    v8f acc[NT] = {};
#pragma unroll 4
    for (int k0 = 0; k0 < K; k0 += 4) {
        const int ka = k0 + half2;
        v2f a;
        a.x = xrow[ka];
        a.y = xrow[ka + 1];
#pragma unroll
        for (int nt = 0; nt < NT; ++nt) {
            const int ncol = nt * 16 + l15;
            v2f b;
            b.x = W[(size_t)ka * NOUT + ncol];
            b.y = W[(size_t)(ka + 1) * NOUT + ncol];
            acc[nt] = __builtin_amdgcn_wmma_f32_16x16x4_f32(
                false, a, false, b, (short)0, acc[nt], false, false);
        }
    }

    const int mbase = mtile * 16 + (lane >> 4) * 8;  // rows j / j+8 split
#pragma unroll
    for (int nt = 0; nt < NT; ++nt) {
        const int ncol = nt * 16 + l15;
        const float bv = BIAS ? bias[ncol] : 0.0f;
#pragma unroll
        for (int j = 0; j < 8; ++j)
            Y[(size_t)(mbase + j) * NOUT + ncol] = acc[nt][j] + bv;
    }
}

// ---------------------------------------------------------------------------
// agg[v][f] = h[v][f] * dinv[v]^2   (self-loop contribution, also inits agg)
__global__ void k_selfloop(float* __restrict__ agg, const float* __restrict__ h,
                           const float* __restrict__ dinv, int total) {
    int i = blockIdx.x * blockDim.x + threadIdx.x;
    if (i < total) {
        float dv = dinv[i >> 7];
        agg[i] = h[i] * dv * dv;
    }
}

// 32 lanes per edge, float4 per lane (128 feats): agg[dst] += h[src]*norm
__global__ void k_scatter128(float* __restrict__ agg, const float* __restrict__ h,
                             const float* __restrict__ dinv,
                             const int* __restrict__ src, const int* __restrict__ dst,
                             const float* __restrict__ ew, int e_cnt) {
    long long idx = (long long)blockIdx.x * blockDim.x + threadIdx.x;
    int e = (int)(idx >> 5);
    int lane = (int)(idx & 31);
    if (e >= e_cnt) return;
    int s = src[e], d = dst[e];
    float norm = dinv[s] * ew[e] * dinv[d];
    float4 v = ((const float4*)(h + (size_t)s * 128))[lane];
    float* ad = agg + (size_t)d * 128 + lane * 4;
    atomicAdd(ad + 0, v.x * norm);
    atomicAdd(ad + 1, v.y * norm);
    atomicAdd(ad + 2, v.z * norm);
    atomicAdd(ad + 3, v.w * norm);
}

// xh = elu(agg + b) in place
__global__ void k_bias_elu(float* __restrict__ agg, const float* __restrict__ b,
                           int total) {
    int i = blockIdx.x * blockDim.x + threadIdx.x;
    if (i < total) {
        float v = agg[i] + b[i & 127];
        agg[i] = (v > 0.0f) ? v : expm1f(v);
    }
}

__global__ void k_zero(float* __restrict__ p, int total) {
    int i = blockIdx.x * blockDim.x + threadIdx.x;
    if (i < total) p[i] = 0.0f;
}

// 32 lanes per edge, float2 per lane (64 feats): bip[bdst] += msg[bsrc]
__global__ void k_scatter64(float* __restrict__ bip, const float* __restrict__ msg,
                            const int* __restrict__ bsrc, const int* __restrict__ bdst,
                            int e_cnt) {
    long long idx = (long long)blockIdx.x * blockDim.x + threadIdx.x;
    int e = (int)(idx >> 5);
    int lane = (int)(idx & 31);
    if (e >= e_cnt) return;
    int s = bsrc[e], d = bdst[e];
    float2 v = ((const float2*)(msg + (size_t)s * 64))[lane];
    float* bd = bip + (size_t)d * 64 + lane * 2;
    atomicAdd(bd + 0, v.x);
    atomicAdd(bd + 1, v.y);
}

// out[r] = elu(bip[r]) @ W_lin + b_lin   (K=64, C=10)
__global__ void k_final(const float* __restrict__ bip, const float* __restrict__ Wl,
                        const float* __restrict__ bl, float* __restrict__ out, int n) {
    int r = blockIdx.x * blockDim.x + threadIdx.x;
    if (r >= n) return;
    float acc[10];
#pragma unroll
    for (int c = 0; c < 10; ++c) acc[c] = bl[c];
    const float* row = bip + (size_t)r * 64;
#pragma unroll 4
    for (int k = 0; k < 64; ++k) {
        float v = row[k];
        v = (v > 0.0f) ? v : expm1f(v);
#pragma unroll
        for (int c = 0; c < 10; ++c) acc[c] += v * Wl[k * 10 + c];
    }
#pragma unroll
    for (int c = 0; c < 10; ++c) out[(size_t)r * 10 + c] = acc[c];
}

// ---------------------------------------------------------------------------
static inline size_t align256(size_t x) { return (x + 255) & ~(size_t)255; }

extern "C" void kernel_launch(void* const* d_in, const int* in_sizes, int n_in,
                              void* d_out, int out_size, void* d_ws, size_t ws_size,
                              hipStream_t stream) {
    // inputs (first-order branch d_in[0,2,3,7,8,13,14] is dead code)
    const float* x_h   = (const float*)d_in[1];
    const int*   eih   = (const int*)d_in[4];    // [2, EH]
    const float* ewh   = (const float*)d_in[5];  // [EH]
    const int*   bidx  = (const int*)d_in[6];    // [2, EB]
    const float* W_ho  = (const float*)d_in[9];
    const float* b_ho  = (const float*)d_in[10];
    const float* W_b1  = (const float*)d_in[11];
    const float* b_b1  = (const float*)d_in[12];
    const float* W_lin = (const float*)d_in[15];
    const float* b_lin = (const float*)d_in[16];
    float* out = (float*)d_out;

    const int F = 128, H1 = 64, C = 10;
    const int NH = in_sizes[1] / F;     // 50000
    const int EH = in_sizes[5];         // 800000
    const int EB = in_sizes[6] / 2;     // 50000
    const int N  = out_size / C;        // 50000

    // workspace layout (peak ~52 MB):
    //   dinv[NH] | h[NH*128] (reused as msg[NH*64]) | agg[NH*128] (reused as bip[N*64])
    char* ws = (char*)d_ws;
    float* dinv = (float*)ws;
    size_t off1 = align256((size_t)NH * 4);
    float* hbuf = (float*)(ws + off1);
    size_t off2 = off1 + align256((size_t)NH * 128 * 4);
    float* agg  = (float*)(ws + off2);
    float* msg  = hbuf;   // h dead after scatter128
    float* bip  = agg;    // agg(xh) dead after gemm2

    const int* eh_src = eih;
    const int* eh_dst = eih + EH;
    const int* b_src  = bidx;
    const int* b_dst  = bidx + EB;

    // 1) degrees + dinv
    k_init_deg<<<(NH + THREADS - 1) / THREADS, THREADS, 0, stream>>>(dinv, NH);
    k_deg_edges<<<(EH + THREADS - 1) / THREADS, THREADS, 0, stream>>>(dinv, eh_dst, ewh, EH);
    k_dinv<<<(NH + THREADS - 1) / THREADS, THREADS, 0, stream>>>(dinv, NH);

    // 2) h = x_h @ W_ho  (WMMA fp32)
    int mtiles = (NH + 15) / 16;
    int gemm_blocks = (mtiles + (THREADS / 32) - 1) / (THREADS / 32);
    k_wmma_gemm_k128<128, false><<<gemm_blocks, THREADS, 0, stream>>>(
        x_h, W_ho, nullptr, hbuf, mtiles);

    // 3) GCN aggregation: self loop init + edge scatter
    int tot128 = NH * 128;
    k_selfloop<<<(tot128 + THREADS - 1) / THREADS, THREADS, 0, stream>>>(
        agg, hbuf, dinv, tot128);
    long long sc_threads = (long long)EH * 32;
    k_scatter128<<<(int)((sc_threads + THREADS - 1) / THREADS), THREADS, 0, stream>>>(
        agg, hbuf, dinv, eh_src, eh_dst, ewh, EH);

    // 4) xh = elu(agg + b_ho)
    k_bias_elu<<<(tot128 + THREADS - 1) / THREADS, THREADS, 0, stream>>>(
        agg, b_ho, tot128);

    // 5) msg = xh @ W_bip1 + b_bip1  (WMMA fp32)
    k_wmma_gemm_k128<64, true><<<gemm_blocks, THREADS, 0, stream>>>(
        agg, W_b1, b_b1, msg, mtiles);

    // 6) bipartite scatter-add
    int tot64 = N * H1;
    k_zero<<<(tot64 + THREADS - 1) / THREADS, THREADS, 0, stream>>>(bip, tot64);
    long long sb_threads = (long long)EB * 32;
    k_scatter64<<<(int)((sb_threads + THREADS - 1) / THREADS), THREADS, 0, stream>>>(
        bip, msg, b_src, b_dst, EB);

    // 7) out = elu(bip) @ W_lin + b_lin
    k_final<<<(N + THREADS - 1) / THREADS, THREADS, 0, stream>>>(
        bip, W_lin, b_lin, out, N);
}